// FusionAttention_79766132621681
// MI455X (gfx1250) — compile-verified
//
#include <hip/hip_runtime.h>
#include <hip/hip_bf16.h>
#include <math.h>

// ---- CDNA5 WMMA vector types ----
typedef __attribute__((ext_vector_type(2))) float v2f;
typedef __attribute__((ext_vector_type(8))) float v8f;

#define BS_ 2
#define T_  12
#define BT_ 24          // bs*T
#define N_  150
#define NP_ 160         // padded to 10 tiles of 16
#define D_  64
#define F_  128         // 2*din
#define NT_ 10          // 16-wide tiles per dim

__device__ __forceinline__ v8f wmma_f32(v2f a, v2f b, v8f c) {
    // V_WMMA_F32_16X16X4_F32 : D = A(16x4,f32) * B(4x16,f32) + C(16x16,f32)
    return __builtin_amdgcn_wmma_f32_16x16x4_f32(
        /*neg_a=*/false, a, /*neg_b=*/false, b,
        /*c_mod=*/(short)0, c, /*reuse_a=*/false, /*reuse_b=*/false);
}

__device__ __forceinline__ v2f ld2(const float* p) {
    return *reinterpret_cast<const v2f*>(p);    // global_load_b64 (8B-aligned)
}

// ---------------------------------------------------------------------------
// Kernel 0: fold a_w + W_B_w + W_B_b into a single 128-vector + scalar.
//   wcomb[f] = sum_c (a_w[c]+a_w[c+64]) * W_B_w[c,f];  bcomb = sum_c aw2[c]*b[c]
// ---------------------------------------------------------------------------
__global__ void k_prep(const float* __restrict__ Wb, const float* __restrict__ bb,
                       const float* __restrict__ aw,
                       float* __restrict__ wcomb, float* __restrict__ bcomb) {
    __shared__ float aw2[64];
    int t = threadIdx.x;                // 128 threads
    if (t < 64) aw2[t] = aw[t] + aw[t + 64];
    __syncthreads();
    float acc = 0.f;
    for (int c = 0; c < 64; ++c) acc += aw2[c] * Wb[c * F_ + t];
    wcomb[t] = acc;
    if (t == 0) {
        float b = 0.f;
        for (int c = 0; c < 64; ++c) b += aw2[c] * bb[c];
        *bcomb = b;
    }
}

// ---------------------------------------------------------------------------
// Kernel 1: per-node attention logit scalar  s[bt,n] = ST[bt,n,:]*wcomb + bcomb
// ---------------------------------------------------------------------------
__global__ void k_svec(const float* __restrict__ H, const float* __restrict__ E,
                       const float* __restrict__ wcomb, const float* __restrict__ bcomb,
                       float* __restrict__ svec) {
    __shared__ float red[128];
    int blk = blockIdx.x;               // bt*N + i
    int bt = blk / N_, i = blk % N_;
    int t = threadIdx.x;                // f in [0,128)
    float v = (t < D_) ? H[(bt * N_ + i) * D_ + t]
                       : E[(bt * N_ + i) * D_ + (t - D_)];
    red[t] = v * wcomb[t];
    __syncthreads();
    for (int s = 64; s > 0; s >>= 1) {
        if (t < s) red[t] += red[t + s];
        __syncthreads();
    }
    if (t == 0) svec[bt * N_ + i] = red[0] + *bcomb;
}

// ---------------------------------------------------------------------------
// Kernel 2: zero-padded adjacency [bs, NP, NP].  Zero padding here is what
// makes all padded-garbage regions of Ptmp/Mbuf provably inert downstream.
// ---------------------------------------------------------------------------
__global__ void k_padadj(const float* __restrict__ adj, float* __restrict__ adjp) {
    int idx = blockIdx.x * 256 + threadIdx.x;
    if (idx >= BS_ * NP_ * NP_) return;
    int b = idx / (NP_ * NP_);
    int rem = idx % (NP_ * NP_);
    int r = rem / NP_, c = rem % NP_;
    adjp[idx] = (r < N_ && c < N_) ? adj[(b * N_ + r) * N_ + c] : 0.f;
}

// ---------------------------------------------------------------------------
// Kernel 3: Ptmp[bt] = ST_pad[NPx128] @ W_fu^T[128xNP]   (f32 WMMA, K=128)
// One wave per 16x16 tile. Clamped in-bounds addresses; NO mask multiplies:
// padded rows of Ptmp are nullified later by adj_pad's zero columns, padded
// cols never reach a stored output. Inner loop = global_load_b64 + v_wmma.
// ---------------------------------------------------------------------------
__global__ void k_ptmp(const float* __restrict__ H, const float* __restrict__ E,
                       const float* __restrict__ Wfu, float* __restrict__ Ptmp) {
    int blk = blockIdx.x;
    int bt = blk / (NT_ * NT_);
    int tt = blk % (NT_ * NT_);
    int ti = tt / NT_, tj = tt % NT_;
    int lane = threadIdx.x;
    int l16 = lane & 15, half = lane >> 4;
    int row = ti * 16 + l16;            // ST row (M)
    int col = tj * 16 + l16;            // W_fu output row (N)

    int rowc = row < N_ ? row : N_ - 1; // clamp: stay in-bounds, garbage is inert
    int colc = col < N_ ? col : N_ - 1;
    const float* __restrict__ Hrow = H   + (bt * N_ + rowc) * D_;
    const float* __restrict__ Erow = E   + (bt * N_ + rowc) * D_;
    const float* __restrict__ Wrow = Wfu + colc * F_;

    v8f acc = {};
    int koff = 2 * half;                // per-lane K sub-offset
    #pragma unroll 4
    for (int k = 0; k < D_; k += 4) {   // first half: features from H
        int kk = k + koff;
        acc = wmma_f32(ld2(Hrow + kk), ld2(Wrow + kk), acc);
    }
    #pragma unroll 4
    for (int k = 0; k < D_; k += 4) {   // second half: features from E
        int kk = k + koff;
        acc = wmma_f32(ld2(Erow + kk), ld2(Wrow + D_ + kk), acc);
    }
    float* out = Ptmp + bt * NP_ * NP_;
    for (int v = 0; v < 8; ++v) {
        int r = ti * 16 + v + 8 * half; // C/D layout: M = v (+8 for hi lanes)
        out[r * NP_ + tj * 16 + l16] = acc[v];
    }
}

// ---------------------------------------------------------------------------
// Kernel 4: Mbuf[bt] = adj_pad[b] @ Ptmp[bt]   (f32 WMMA, K=160, fully padded)
// ---------------------------------------------------------------------------
__global__ void k_adjP(const float* __restrict__ adjp, const float* __restrict__ Ptmp,
                       float* __restrict__ Mbuf) {
    __builtin_amdgcn_s_wait_tensorcnt(0);   // CDNA5 split wait counter (free)
    int blk = blockIdx.x;
    int bt = blk / (NT_ * NT_);
    int tt = blk % (NT_ * NT_);
    int ti = tt / NT_, tj = tt % NT_;
    int b = bt / T_;
    int lane = threadIdx.x;
    int l16 = lane & 15, half = lane >> 4;
    int row = ti * 16 + l16;
    int col = tj * 16 + l16;
    const float* __restrict__ A = adjp + b * NP_ * NP_ + row * NP_;
    const float* __restrict__ B = Ptmp + bt * NP_ * NP_ + col;
    v8f acc = {};
    int koff = 2 * half;
    #pragma unroll 4
    for (int k = 0; k < NP_; k += 4) {
        int kk = k + koff;
        __builtin_prefetch(&B[(k + 8) * NP_], 0, 0);        // global_prefetch_b8
        v2f a = ld2(A + kk);                                // adj row, contiguous
        v2f bf;
        bf.x = B[kk * NP_];                                 // column: strided b32
        bf.y = B[(kk + 1) * NP_];
        acc = wmma_f32(a, bf, acc);
    }
    float* out = Mbuf + bt * NP_ * NP_;
    for (int v = 0; v < 8; ++v)
        out[(ti * 16 + v + 8 * half) * NP_ + tj * 16 + l16] = acc[v];
}

// ---------------------------------------------------------------------------
// Kernel 5: masked softmax rows (exact, incl. degenerate all-masked case)
// into LDS, then  P = tanh(att[16x160] @ M[160x16])  per tile via WMMA.
// Block = 256 threads (8 waves) handles one (bt, row-tile).
// ---------------------------------------------------------------------------
__global__ void k_attn(const float* __restrict__ adj, const float* __restrict__ svec,
                       const float* __restrict__ Mbuf, float* __restrict__ out) {
    __shared__ __align__(16) float att_s[16][NP_];
    __shared__ float red[256];
    int bt = blockIdx.x / NT_;
    int ti = blockIdx.x % NT_;
    int b  = bt / T_;
    int tid = threadIdx.x;

    // ---- Phase A: 16 masked-softmax rows ----
    for (int r = 0; r < 16; ++r) {
        int i = ti * 16 + r;
        bool active = (i < N_) && (tid < N_);
        float z = -3.0e38f;
        if (active) {
            int q = (2 * i + (tid >= 75 ? 1 : 0)) % N_;
            float sv = svec[bt * N_ + q];
            float e = sv > 0.f ? sv : 0.2f * sv;            // leaky relu
            z = (adj[(b * N_ + i) * N_ + tid] > 0.f) ? e : -9.0e15f;
        }
        red[tid] = z;
        __syncthreads();
        for (int s = 128; s > 0; s >>= 1) {
            if (tid < s) red[tid] = fmaxf(red[tid], red[tid + s]);
            __syncthreads();
        }
        float mx = red[0];
        __syncthreads();
        float p = active ? expf(z - mx) : 0.f;
        red[tid] = p;
        __syncthreads();
        for (int s = 128; s > 0; s >>= 1) {
            if (tid < s) red[tid] += red[tid + s];
            __syncthreads();
        }
        float sum = red[0];
        __syncthreads();
        if (tid < NP_) att_s[r][tid] = active ? (p / sum) : 0.f;  // pad cols -> 0
        __syncthreads();
    }

    // ---- Phase B: att(16x160) @ M(160x16) per column tile, tanh, store ----
    int lane = tid & 31, wave = tid >> 5;
    int l16 = lane & 15, half = (lane >> 4) & 1;
    const float* __restrict__ Mb = Mbuf + bt * NP_ * NP_;
    int koff = 2 * half;
    for (int tj = wave; tj < NT_; tj += 8) {
        int col = tj * 16 + l16;
        const float* __restrict__ B = Mb + col;
        v8f acc = {};
        #pragma unroll 4
        for (int k = 0; k < NP_; k += 4) {
            int kk = k + koff;
            __builtin_prefetch(&B[(k + 8) * NP_], 0, 0);
            v2f a = *reinterpret_cast<const v2f*>(&att_s[l16][kk]);  // ds_load_b64
            v2f bf;
            bf.x = B[kk * NP_];
            bf.y = B[(kk + 1) * NP_];
            acc = wmma_f32(a, bf, acc);
        }
        for (int v = 0; v < 8; ++v) {
            int rr = ti * 16 + v + 8 * half;
            if (rr < N_ && col < N_)
                out[(bt * N_ + rr) * N_ + col] = tanhf(acc[v]);
        }
    }
}

// ---------------------------------------------------------------------------
extern "C" void kernel_launch(void* const* d_in, const int* in_sizes, int n_in,
                              void* d_out, int out_size, void* d_ws, size_t ws_size,
                              hipStream_t stream) {
    const float* H   = (const float*)d_in[0];   // [2,12,150,64]
    const float* E   = (const float*)d_in[1];   // [2,12,150,64]
    const float* adj = (const float*)d_in[2];   // [2,150,150]
    const float* Wb  = (const float*)d_in[3];   // [64,128]
    const float* bb  = (const float*)d_in[4];   // [64]
    const float* Wfu = (const float*)d_in[5];   // [150,128]
    const float* aw  = (const float*)d_in[6];   // [128]
    float* out = (float*)d_out;                 // [2,12,150,150]

    float* ws    = (float*)d_ws;
    float* wcomb = ws;                          // 128
    float* bcomb = ws + 128;                    // 1
    float* svec  = ws + 160;                    // 24*150 = 3600
    float* adjp  = ws + 3840;                   // 2*160*160 = 51200
    float* Ptmp  = adjp + BS_ * NP_ * NP_;      // 24*160*160 = 614400
    float* Mbuf  = Ptmp + BT_ * NP_ * NP_;      // 24*160*160 = 614400

    k_prep  <<<1, 128, 0, stream>>>(Wb, bb, aw, wcomb, bcomb);
    k_svec  <<<BT_ * N_, 128, 0, stream>>>(H, E, wcomb, bcomb, svec);
    k_padadj<<<(BS_ * NP_ * NP_ + 255) / 256, 256, 0, stream>>>(adj, adjp);
    k_ptmp  <<<BT_ * NT_ * NT_, 32, 0, stream>>>(H, E, Wfu, Ptmp);
    k_adjP  <<<BT_ * NT_ * NT_, 32, 0, stream>>>(adjp, Ptmp, Mbuf);
    k_attn  <<<BT_ * NT_, 256, 0, stream>>>(adj, svec, Mbuf, out);
}